// GNNRecommender_19731079758363
// MI455X (gfx1250) — compile-verified
//
#include <hip/hip_runtime.h>
#include <hip/hip_bf16.h>

// ---------------------------------------------------------------------------
// GNN recommender: proj GEMM (WMMA f32) -> 3x (GEMM + edge scatter + finalize)
// Target: gfx1250 (MI455X), wave32, V_WMMA_F32_16X16X4_F32
// B tile staged in LDS in WMMA *fragment order*; inner loop batches all
// B fragments of a k4-step into registers, then bursts the WMMAs.
// ---------------------------------------------------------------------------

#define NUM_USERS 1024
#define NUM_NODES 101024      // 1024 + 100000
#define NUM_EDGES 2000000
#define FEAT_DIM  1024

typedef float v2f __attribute__((ext_vector_type(2)));
typedef float v8f __attribute__((ext_vector_type(8)));

// ---------------------------------------------------------------------------
// degree / norm kernels
// ---------------------------------------------------------------------------
__global__ void init_deg_kernel(float* __restrict__ deg, int n) {
    int i = blockIdx.x * blockDim.x + threadIdx.x;
    if (i < n) deg[i] = 1.0f;   // self loop
}

__global__ void add_deg_kernel(const int* __restrict__ dst, float* __restrict__ deg, int e) {
    int i = blockIdx.x * blockDim.x + threadIdx.x;
    if (i < e) atomicAdd(&deg[dst[i]], 1.0f);
}

__global__ void make_dinv_kernel(float* __restrict__ deg, int n) {
    int i = blockIdx.x * blockDim.x + threadIdx.x;
    if (i < n) {
        float d = deg[i];
        deg[i] = (d > 0.0f) ? rsqrtf(d) : 0.0f;
    }
}

__global__ void zero_kernel(float* __restrict__ p, long n) {
    long i = (long)blockIdx.x * blockDim.x + threadIdx.x;
    if (i < n) p[i] = 0.0f;
}

// ---------------------------------------------------------------------------
// Tiled fp32 WMMA GEMM:  Y[M x OC] = X[M x K] @ W[K x OC] (+ bias)
// Block: 256 threads = 8 waves; block tile = 128 rows x OC cols; K chunk = 64.
// Wave w computes rows [w*16, w*16+16) x all OC cols -> OC/16 v8f accumulators.
// Rows < splitRow use (Wa, biasA); rows >= splitRow use (Wb, biasB).
//
// LDS layouts:
//   As[row][k]   row-major, +4 pad -> A-frag read = aligned ds_load_b64,
//                bank pairs 4m..4m+1 (lo half) / 4m+2..4m+3 (hi half): no conflicts.
//   Bs           fragment order: float2 per lane per (k4-block kb, col tile ct):
//                idx = ((kb*NCT + ct)*32 + lane)*2 + p
//                lane = (kr>=2)*16 + (n&15), p = kr&1, kb=k/4, kr=k%4, ct=n/16.
//                B-frag read = ds_load_b64, lane L owns banks 2L,2L+1: no conflicts.
// ---------------------------------------------------------------------------
template <int OC>
__global__ __launch_bounds__(256) void gemm_wmma_kernel(
    const float* __restrict__ X, int M, int K,
    const float* __restrict__ Wa, const float* __restrict__ Wb,
    const float* __restrict__ biasA, const float* __restrict__ biasB,
    int splitRow,
    float* __restrict__ Y)
{
    constexpr int TM = 128;
    constexpr int TK = 64;
    constexpr int APAD = 4;
    constexpr int NCT = OC / 16;

    __shared__ float As[TM][TK + APAD];
    __shared__ float Bs[(TK / 4) * NCT * 64];   // fragment-ordered B tile

    const int r0   = blockIdx.x * TM;
    const float* W    = (r0 < splitRow) ? Wa : Wb;
    const float* bias = (r0 < splitRow) ? biasA : biasB;

    const int tid  = threadIdx.x;
    const int lane = tid & 31;
    const int wave = tid >> 5;              // 0..7

    v8f acc[NCT] = {};

    const int rowBase = wave * 16;
    const int m    = lane & 15;             // row-in-tile (A) / col-in-tile (D)
    const int kOff = (lane >> 4) << 1;      // 0 for lanes 0-15, 2 for lanes 16-31

    for (int kc = 0; kc < K; kc += TK) {
        // ---- stage A tile: 128 x 64 floats (2048 float4, 8 per thread) ----
        #pragma unroll
        for (int c = 0; c < (TM * TK) / (256 * 4); ++c) {
            int idx = c * 256 + tid;               // float4 index
            int row = idx / (TK / 4);
            int c4  = idx % (TK / 4);
            int gr  = r0 + row;
            if (gr >= M) gr = M - 1;               // clamp (stores are guarded)
            const float4 v = *reinterpret_cast<const float4*>(
                X + (long)gr * K + kc + c4 * 4);
            *reinterpret_cast<float4*>(&As[row][c4 * 4]) = v;
        }
        // ---- stage B tile (TK x OC) directly into fragment order ----
        #pragma unroll
        for (int c = 0; c < (TK * OC) / (256 * 4); ++c) {
            int idx = c * 256 + tid;
            int k   = idx / (OC / 4);
            int c4  = idx % (OC / 4);
            const float4 v = *reinterpret_cast<const float4*>(
                W + (long)(kc + k) * OC + c4 * 4);
            const int kb = k >> 2;
            const int kr = k & 3;
            const int p  = kr & 1;
            const int laneHi = (kr >> 1) << 4;
            #pragma unroll
            for (int j = 0; j < 4; ++j) {
                int n  = c4 * 4 + j;
                int ct = n >> 4;
                int nl = n & 15;
                Bs[((kb * NCT + ct) * 32 + laneHi + nl) * 2 + p] =
                    reinterpret_cast<const float*>(&v)[j];
            }
        }

        // ---- software prefetch of next K chunk (global_prefetch_b8) ----
        if (kc + TK < K) {
            // A: 128 rows x 64 floats = 256 cachelines; 1 per thread
            int prow = tid >> 1, phalf = tid & 1;
            int pgr = r0 + prow; if (pgr >= M) pgr = M - 1;
            __builtin_prefetch(X + (long)pgr * K + (kc + TK) + phalf * 32, 0, 1);
            // B: TK rows x OC floats
            int brow = tid >> 2, bq = tid & 3;
            if (bq * 32 < OC) {
                __builtin_prefetch(W + (long)(kc + TK + brow) * OC + bq * 32, 0, 1);
            }
        }
        __syncthreads();

        // ---- 16 WMMA k-steps over this chunk ----
        // Per k4-step: load A frag + ALL B frags into registers, then burst
        // the NCT WMMAs back-to-back (single ds wait covers all loads).
        #pragma unroll 4
        for (int kk4 = 0; kk4 < TK / 4; ++kk4) {
            v2f a = *reinterpret_cast<const v2f*>(&As[rowBase + m][kk4 * 4 + kOff]);
            const v2f* bfr = reinterpret_cast<const v2f*>(&Bs[kk4 * NCT * 64]) + lane;
            v2f bfrag[NCT];
            #pragma unroll
            for (int ct = 0; ct < NCT; ++ct) {
                bfrag[ct] = bfr[ct * 32];
            }
            #pragma unroll
            for (int ct = 0; ct < NCT; ++ct) {
                acc[ct] = __builtin_amdgcn_wmma_f32_16x16x4_f32(
                    /*neg_a=*/false, a, /*neg_b=*/false, bfrag[ct],
                    /*c_mod=*/(short)0, acc[ct],
                    /*reuse_a=*/false, /*reuse_b=*/false);
            }
        }
        __syncthreads();
    }

    // ---- store D: VGPR r -> row r (lanes 0-15) / row r+8 (lanes 16-31) ----
    const int rAdd = (lane >> 4) * 8;
    #pragma unroll
    for (int ct = 0; ct < NCT; ++ct) {
        const int col = ct * 16 + m;
        const float bv = bias ? bias[col] : 0.0f;
        #pragma unroll
        for (int r = 0; r < 8; ++r) {
            int row = r0 + rowBase + rAdd + r;
            if (row < M) {
                Y[(long)row * OC + col] = acc[ct][r] + bv;
            }
        }
    }
}

// ---------------------------------------------------------------------------
// Edge scatter: one wave per edge; acc[dst] += hw[src] * dinv[src]*dinv[dst]
// ---------------------------------------------------------------------------
template <int OC>
__global__ __launch_bounds__(256) void scatter_kernel(
    const int* __restrict__ src, const int* __restrict__ dst,
    const float* __restrict__ dinv, const float* __restrict__ hw,
    float* __restrict__ acc, int e)
{
    int wid  = (int)(((long)blockIdx.x * blockDim.x + threadIdx.x) >> 5);
    int lane = threadIdx.x & 31;
    if (wid >= e) return;
    int s = src[wid];
    int d = dst[wid];
    float norm = dinv[s] * dinv[d];
    const float* hs = hw + (long)s * OC;
    float* ad = acc + (long)d * OC;
    #pragma unroll
    for (int c = lane; c < OC; c += 32) {
        atomicAdd(&ad[c], hs[c] * norm);
    }
}

// ---------------------------------------------------------------------------
// Finalize: out = (acc + hw*dinv^2 [self loop] + bias), optional ReLU
// ---------------------------------------------------------------------------
template <int OC, bool RELU>
__global__ void finalize_kernel(
    const float* __restrict__ acc, const float* __restrict__ hw,
    const float* __restrict__ dinv, const float* __restrict__ bias,
    float* __restrict__ out, int nNodes)
{
    long i = (long)blockIdx.x * blockDim.x + threadIdx.x;
    if (i >= (long)nNodes * OC) return;
    int node = (int)(i / OC);
    int c    = (int)(i % OC);
    float di = dinv[node];
    float v = acc[i] + hw[i] * di * di + bias[c];
    if (RELU) v = fmaxf(v, 0.0f);
    out[i] = v;
}

// ---------------------------------------------------------------------------
// Host orchestration
// ---------------------------------------------------------------------------
extern "C" void kernel_launch(void* const* d_in, const int* in_sizes, int n_in,
                              void* d_out, int out_size, void* d_ws, size_t ws_size,
                              hipStream_t stream) {
    const float* x  = (const float*)d_in[0];
    const int*   ei = (const int*)d_in[1];
    const float* Wu = (const float*)d_in[2];
    const float* bu = (const float*)d_in[3];
    const float* Wb = (const float*)d_in[4];
    const float* bb = (const float*)d_in[5];
    const float* W1 = (const float*)d_in[6];
    const float* b1 = (const float*)d_in[7];
    const float* W2 = (const float*)d_in[8];
    const float* b2 = (const float*)d_in[9];
    const float* W3 = (const float*)d_in[10];
    const float* b3 = (const float*)d_in[11];
    float* out = (float*)d_out;

    const int N = NUM_NODES;
    const int E = NUM_EDGES;

    const int* srcE = ei;        // edge_index[0, :]
    const int* dstE = ei + E;    // edge_index[1, :]

    // workspace layout (floats)
    float* ws   = (float*)d_ws;
    float* dinv = ws;                                  // N (padded to 101376)
    float* h    = ws + 101376;                         // N*128
    float* hw   = h + (long)N * 128;                   // N*128
    float* acc  = hw + (long)N * 128;                  // N*128

    const int T = 256;
    const int gN   = (N + T - 1) / T;
    const int gE   = (E + T - 1) / T;
    const int gRow = (N + 127) / 128;                  // 790 row blocks
    const int gScat = (E + 7) / 8;                     // 8 waves (edges) per block

    // --- gcn_norm: deg = 1 + sum over incoming edges; dinv = rsqrt(deg) ---
    init_deg_kernel<<<gN, T, 0, stream>>>(dinv, N);
    add_deg_kernel<<<gE, T, 0, stream>>>(dstE, dinv, E);
    make_dinv_kernel<<<gN, T, 0, stream>>>(dinv, N);

    // --- input projection: h = x @ (Wu|Wb) + (bu|bb) ---
    gemm_wmma_kernel<128><<<gRow, T, 0, stream>>>(
        x, N, FEAT_DIM, Wu, Wb, bu, bb, NUM_USERS, h);

    // --- layer 1: relu(gcn_conv(h, W1, b1)) ---
    gemm_wmma_kernel<128><<<gRow, T, 0, stream>>>(
        h, N, 128, W1, W1, nullptr, nullptr, N, hw);
    zero_kernel<<<(int)(((long)N * 128 + T - 1) / T), T, 0, stream>>>(acc, (long)N * 128);
    scatter_kernel<128><<<gScat, T, 0, stream>>>(srcE, dstE, dinv, hw, acc, E);
    finalize_kernel<128, true><<<(int)(((long)N * 128 + T - 1) / T), T, 0, stream>>>(
        acc, hw, dinv, b1, h, N);

    // --- layer 2: relu(gcn_conv(h, W2, b2)) ---
    gemm_wmma_kernel<128><<<gRow, T, 0, stream>>>(
        h, N, 128, W2, W2, nullptr, nullptr, N, hw);
    zero_kernel<<<(int)(((long)N * 128 + T - 1) / T), T, 0, stream>>>(acc, (long)N * 128);
    scatter_kernel<128><<<gScat, T, 0, stream>>>(srcE, dstE, dinv, hw, acc, E);
    finalize_kernel<128, true><<<(int)(((long)N * 128 + T - 1) / T), T, 0, stream>>>(
        acc, hw, dinv, b2, h, N);

    // --- layer 3: gcn_conv(h, W3, b3), no relu, write d_out ---
    gemm_wmma_kernel<64><<<gRow, T, 0, stream>>>(
        h, N, 128, W3, W3, nullptr, nullptr, N, hw);
    zero_kernel<<<(int)(((long)N * 64 + T - 1) / T), T, 0, stream>>>(acc, (long)N * 64);
    scatter_kernel<64><<<gScat, T, 0, stream>>>(srcE, dstE, dinv, hw, acc, E);
    finalize_kernel<64, false><<<(int)(((long)N * 64 + T - 1) / T), T, 0, stream>>>(
        acc, hw, dinv, b3, out, N);
}